// ONINorm_31267361915441
// MI455X (gfx1250) — compile-verified
//
#include <hip/hip_runtime.h>

// ONINorm for (262144, 256) f32, NORM_GROUPS=8 -> per group: (n=262144, c=32).
// Memory-bound target: 768 MB total traffic ~= 33 us @ 23.3 TB/s.
// Both tall-skinny GEMMs (covariance Z^T Z and apply Zc*B^T) use
// V_WMMA_F32_16X16X4_F32 (full f32 precision matrix pipe).

#define GROUPS 8
#define CDIM   32
#define NPG    262144           // rows per group
#define EPSF   1e-5f

typedef float v2f __attribute__((ext_vector_type(2)));
typedef float v8f __attribute__((ext_vector_type(8)));

static __device__ __forceinline__ v8f vzero8() {
    v8f z = {0.f, 0.f, 0.f, 0.f, 0.f, 0.f, 0.f, 0.f};
    return z;
}

// D(16x16,f32) = A(16x4,f32) x B(4x16,f32) + C
static __device__ __forceinline__ v8f wmma4(v2f a, v2f b, v8f c) {
    return __builtin_amdgcn_wmma_f32_16x16x4_f32(
        /*neg_a=*/false, a, /*neg_b=*/false, b,
        /*c_mod=*/(short)0, c, /*reuse_a=*/false, /*reuse_b=*/false);
}

// ---------------------------------------------------------------------------
// Kernel 1: per-group sum(Z) and Gram matrix sum(Z Z^T) partials.
// 1024 blocks x 256 threads. Block b -> group b/128, 8 iterations x 256 rows.
// Register-prefetch pipeline overlaps global loads with the WMMA K-loop;
// channel sums ride along in registers (fixed channel phase per thread).
// launch_bounds(...,4) keeps VGPRs <= 256 (no s_set_vgpr_msb, multi-block
// residency per WGP for latency hiding); unroll 2 caps staged LDS operands.
// ---------------------------------------------------------------------------
__global__ __launch_bounds__(256, 4) void k_cov(const float* __restrict__ in,
                                                float* __restrict__ wsum,
                                                float* __restrict__ wS) {
    __shared__ float4 smem4[2048];           // 32 KB: 256 rows x 32 ch
    float* smem = (float*)smem4;

    const int tid  = threadIdx.x;
    const int lane = tid & 31;
    const int wv   = tid >> 5;
    const int b    = blockIdx.x;
    const int g    = b >> 7;
    const int lb   = b & 127;
    const int half = lane >> 4;              // 0|1
    const int l15  = lane & 15;

    v8f c00 = vzero8(), c01 = vzero8(), c10 = vzero8(), c11 = vzero8();
    float sz0 = 0.f, sz1 = 0.f, sz2 = 0.f, sz3 = 0.f;

    const float4* src =
        (const float4*)(in + ((long)g * NPG + (long)lb * 2048) * CDIM);

    float4 r[8];
    #pragma unroll
    for (int i = 0; i < 8; ++i) r[i] = src[i * 256 + tid];

    for (int it = 0; it < 8; ++it) {
        // channel partial sums from the prefetch registers (thread's float4
        // always covers channels (tid*4)&31 .. +3 since 1024 % 32 == 0).
        #pragma unroll
        for (int i = 0; i < 8; ++i) {
            sz0 += r[i].x; sz1 += r[i].y; sz2 += r[i].z; sz3 += r[i].w;
        }
        __syncthreads();                     // prior iter done reading LDS
        #pragma unroll
        for (int i = 0; i < 8; ++i) smem4[i * 256 + tid] = r[i];
        __syncthreads();
        if (it < 7) {                        // prefetch next tile (overlaps WMMA)
            #pragma unroll
            for (int i = 0; i < 8; ++i)
                r[i] = src[(it + 1) * 2048 + i * 256 + tid];
        }

        // Gram matrix: 8 K-steps of 4 rows over this wave's 32 rows.
        // A and B operands are identical staging functions (symmetric GEMM):
        // P(ct)[lane] = Z[r0 + 2*half + {0,1}][16*ct + l15].
        const int rbase = wv * 32;
        #pragma unroll 2
        for (int s = 0; s < 8; ++s) {
            const int rr = rbase + s * 4 + 2 * half;
            v2f p0, p1;
            p0.x = smem[rr * CDIM + l15];
            p0.y = smem[(rr + 1) * CDIM + l15];
            p1.x = smem[rr * CDIM + 16 + l15];
            p1.y = smem[(rr + 1) * CDIM + 16 + l15];
            c00 = wmma4(p0, p0, c00);
            c01 = wmma4(p0, p1, c01);
            c10 = wmma4(p1, p0, c10);
            c11 = wmma4(p1, p1, c11);
        }
    }

    // Block-level reduction: reuse tile LDS, [0..1024)=S, [1024..1056)=sum.
    __syncthreads();
    float* acc = smem;
    for (int i = tid; i < 1056; i += 256) acc[i] = 0.f;
    __syncthreads();

    // C/D layout: lane L, VGPR v -> (M = v + 8*(L>=16), N = L&15)
    #pragma unroll
    for (int v = 0; v < 8; ++v) {
        const int m = v + 8 * half;
        atomicAdd(&acc[m * CDIM + l15],             c00[v]);
        atomicAdd(&acc[m * CDIM + 16 + l15],        c01[v]);
        atomicAdd(&acc[(16 + m) * CDIM + l15],      c10[v]);
        atomicAdd(&acc[(16 + m) * CDIM + 16 + l15], c11[v]);
    }
    const int c0 = (tid * 4) & 31;
    atomicAdd(&acc[1024 + c0 + 0], sz0);
    atomicAdd(&acc[1024 + c0 + 1], sz1);
    atomicAdd(&acc[1024 + c0 + 2], sz2);
    atomicAdd(&acc[1024 + c0 + 3], sz3);
    __syncthreads();

    for (int i = tid; i < 1056; i += 256) {
        const float val = acc[i];
        if (i < 1024) unsafeAtomicAdd(&wS[g * 1024 + i], val);
        else          unsafeAtomicAdd(&wsum[g * CDIM + (i - 1024)], val);
    }
}

// ---------------------------------------------------------------------------
// Kernel 2: per-group Newton-Schulz on 32x32 matrices (WMMA matmuls).
// 8 blocks x 32 threads (one wave per group).
// ---------------------------------------------------------------------------
static __device__ void mm32(float* D, const float* A, const float* Bm, int lane) {
    const int half = lane >> 4;
    const int l15  = lane & 15;
    #pragma unroll
    for (int mt = 0; mt < 2; ++mt)
        #pragma unroll
        for (int nt = 0; nt < 2; ++nt) {
            v8f c = vzero8();
            #pragma unroll
            for (int s = 0; s < 8; ++s) {
                const int kb = 4 * s + 2 * half;
                v2f a = *(const v2f*)&A[(16 * mt + l15) * CDIM + kb];
                v2f b;
                b.x = Bm[kb * CDIM + 16 * nt + l15];
                b.y = Bm[(kb + 1) * CDIM + 16 * nt + l15];
                c = wmma4(a, b, c);
            }
            #pragma unroll
            for (int v = 0; v < 8; ++v)
                D[(16 * mt + v + 8 * half) * CDIM + 16 * nt + l15] = c[v];
        }
}

__global__ __launch_bounds__(32) void k_ns(const float* __restrict__ wsum,
                                           const float* __restrict__ wS,
                                           float* __restrict__ wBT,
                                           float* __restrict__ wbias) {
    __shared__ float Smat[1024], Bmat[1024], T1[1024], T2[1024], muS[32];
    const int lane = threadIdx.x;
    const int g    = blockIdx.x;

    const float mu = wsum[g * CDIM + lane] * (1.0f / (float)NPG);
    muS[lane] = mu;
    __syncthreads();

    // Centered covariance: S = Ssum - n*mu*mu^T + eps*I ; Frobenius norm.
    float fro = 0.f;
    for (int i = 0; i < 32; ++i) {
        float v = wS[g * 1024 + i * CDIM + lane]
                - (float)NPG * muS[i] * mu
                + ((i == lane) ? EPSF : 0.f);
        Smat[i * CDIM + lane] = v;
        fro += v * v;
    }
    for (int off = 16; off > 0; off >>= 1) fro += __shfl_xor(fro, off, 32);
    const float norm  = sqrtf(fro);
    const float rnorm = 1.0f / norm;
    for (int i = 0; i < 32; ++i) Smat[i * CDIM + lane] *= rnorm;
    for (int i = 0; i < 32; ++i) Bmat[i * CDIM + lane] = (i == lane) ? 1.f : 0.f;
    __syncthreads();

    // B = 1.5 B - 0.5 (B B B) S, five iterations.
    for (int t = 0; t < 5; ++t) {
        mm32(T1, Bmat, Bmat, lane);
        mm32(T2, T1, Bmat, lane);
        mm32(T1, T2, Smat, lane);
        for (int i = 0; i < 32; ++i)
            Bmat[i * CDIM + lane] = 1.5f * Bmat[i * CDIM + lane]
                                  - 0.5f * T1[i * CDIM + lane];
        __syncthreads();
    }

    // out[m][c] = sum_k Zc[m][k] * Beff[k][c], Beff[k][c] = B[c][k]/sqrt(norm).
    // Store BeffT[c][k] = B[c][k]*rsn (== Bmat row-major copy) + folded bias.
    const float rsn = rsqrtf(norm);
    for (int i = 0; i < 32; ++i)
        wBT[g * 1024 + i * CDIM + lane] = Bmat[i * CDIM + lane] * rsn;
    float bs = 0.f;
    for (int d = 0; d < 32; ++d) bs += muS[d] * Bmat[lane * CDIM + d];
    wbias[g * CDIM + lane] = bs * rsn;
}

// ---------------------------------------------------------------------------
// Kernel 3: out = Z * Beff - bias. 1024 blocks x 256 threads; each wave owns
// 16 tiles of 16 rows; B operands resident in 32 VGPRs.
// ---------------------------------------------------------------------------
__global__ __launch_bounds__(256, 4) void k_apply(const float* __restrict__ in,
                                                  const float* __restrict__ wBT,
                                                  const float* __restrict__ wbias,
                                                  float* __restrict__ out) {
    const int tid  = threadIdx.x;
    const int lane = tid & 31;
    const int wv   = tid >> 5;
    const long w   = (long)blockIdx.x * 8 + wv;      // 0..8191
    const int  g   = (int)(w >> 10);
    const int half = lane >> 4;
    const int l15  = lane & 15;

    v2f  bm[2][8];
    float bv[2];
    #pragma unroll
    for (int nt = 0; nt < 2; ++nt) {
        #pragma unroll
        for (int s = 0; s < 8; ++s)
            bm[nt][s] = *(const v2f*)&wBT[g * 1024 + (16 * nt + l15) * CDIM
                                          + 4 * s + 2 * half];
        bv[nt] = wbias[g * CDIM + 16 * nt + l15];
    }

    for (int t = 0; t < 16; ++t) {
        const long T = w * 16 + t;                   // 16-row tile index
        const float* base = in + T * 512;
        v8f c0 = vzero8(), c1 = vzero8();
        #pragma unroll
        for (int s = 0; s < 8; ++s) {
            v2f a = *(const v2f*)&base[l15 * CDIM + 4 * s + 2 * half];
            c0 = wmma4(a, bm[0][s], c0);
            c1 = wmma4(a, bm[1][s], c1);
        }
        float* ob = out + T * 512;
        #pragma unroll
        for (int v = 0; v < 8; ++v) {
            const int m = v + 8 * half;
            ob[m * CDIM + l15]      = c0[v] - bv[0];
            ob[m * CDIM + 16 + l15] = c1[v] - bv[1];
        }
    }
}

// ---------------------------------------------------------------------------
extern "C" void kernel_launch(void* const* d_in, const int* in_sizes, int n_in,
                              void* d_out, int out_size, void* d_ws, size_t ws_size,
                              hipStream_t stream) {
    const float* in  = (const float*)d_in[0];
    float*       out = (float*)d_out;
    float*       ws  = (float*)d_ws;

    // ws layout (floats): [0,256) sumZ | [256,8448) Ssum | [8448,16640) BeffT
    //                     | [16640,16896) bias
    float* wsum  = ws;
    float* wS    = ws + 256;
    float* wBT   = ws + 8448;
    float* wbias = ws + 16640;

    hipMemsetAsync(ws, 0, 8448 * sizeof(float), stream);   // zero accumulators
    k_cov  <<<1024, 256, 0, stream>>>(in, wsum, wS);
    k_ns   <<<GROUPS, 32, 0, stream>>>(wsum, wS, wBT, wbias);
    k_apply<<<1024, 256, 0, stream>>>(in, wBT, wbias, out);
}